// ElasticGNN_59021440582159
// MI455X (gfx1250) — compile-verified
//
#include <hip/hip_runtime.h>
#include <math.h>

#define IN_F   256
#define HID_F  64
#define OUT_F  32
#define LAMBDA1_C 3.0f
#define GAMMA_C   0.25f   // 1/(1+LAMBDA2), LAMBDA2=3
#define BETA_C    2.0f    // 1/(2*GAMMA)

typedef float v2f __attribute__((ext_vector_type(2)));
typedef float v8f __attribute__((ext_vector_type(8)));

// ---------------------------------------------------------------------------
// Fused MLP encoder: hh = relu(feat@W1 + b1) @ W2 + b2, via f32 WMMA 16x16x4.
// 256 threads = 8 waves; each wave owns a 32-row M tile (block = 256 rows).
// W1^T (64KB) + W2^T (8KB) staged transposed in LDS so every B fragment is a
// single aligned ds_load_b64; per-wave 32x64 h staging (64KB). 136KB dyn LDS.
// ---------------------------------------------------------------------------
__global__ void __launch_bounds__(256) mlp_wmma(
    const float* __restrict__ feat, const float* __restrict__ W1,
    const float* __restrict__ b1,   const float* __restrict__ W2,
    const float* __restrict__ b2,   float* __restrict__ hh, int N)
{
    extern __shared__ float lds[];
    float* ldsW1t = lds;                          // [HID_F][IN_F]  (transposed)
    float* ldsW2t = ldsW1t + IN_F * HID_F;        // [OUT_F][HID_F] (transposed)
    float* ldsH   = ldsW2t + HID_F * OUT_F;       // 8 waves * 32 * HID_F

    const int tid = threadIdx.x;
    // Stage W1 transposed: coalesced float4 global read (4 consecutive n, one k),
    // 4 scalar LDS writes each.
    for (int idx = tid; idx < (IN_F * HID_F) / 4; idx += 256) {
        int k = idx / (HID_F / 4);
        int n4 = (idx % (HID_F / 4)) * 4;
        float4 w = ((const float4*)W1)[idx];
        ldsW1t[(n4 + 0) * IN_F + k] = w.x;
        ldsW1t[(n4 + 1) * IN_F + k] = w.y;
        ldsW1t[(n4 + 2) * IN_F + k] = w.z;
        ldsW1t[(n4 + 3) * IN_F + k] = w.w;
    }
    for (int idx = tid; idx < (HID_F * OUT_F) / 4; idx += 256) {
        int k = idx / (OUT_F / 4);
        int n4 = (idx % (OUT_F / 4)) * 4;
        float4 w = ((const float4*)W2)[idx];
        ldsW2t[(n4 + 0) * HID_F + k] = w.x;
        ldsW2t[(n4 + 1) * HID_F + k] = w.y;
        ldsW2t[(n4 + 2) * HID_F + k] = w.z;
        ldsW2t[(n4 + 3) * HID_F + k] = w.w;
    }
    __syncthreads();

    const int wave = tid >> 5;
    const int lane = tid & 31;
    const int lo = lane & 15;     // row (A,M) / col (B/D, N) within tile
    const int hi = lane >> 4;     // K-pair / M-half selector
    const int row0 = blockIdx.x * 256 + wave * 32;

    // Clamp row indices so A loads are branch-free; out-of-range rows compute
    // garbage that is never stored (stores are guarded by rr < N).
    int r0 = row0 + lo;       if (r0 >= N) r0 = N - 1;
    int r1 = row0 + 16 + lo;  if (r1 >= N) r1 = N - 1;
    const float* frow0 = feat + (size_t)r0 * IN_F;
    const float* frow1 = feat + (size_t)r1 * IN_F;

    // ---- GEMM1: [32 x 256] @ [256 x 64] -> 2 M-tiles x 4 N-tiles of 16x16
    v8f acc[2][4];
#pragma unroll
    for (int mt = 0; mt < 2; ++mt)
#pragma unroll
        for (int nt = 0; nt < 4; ++nt) acc[mt][nt] = v8f{0};

    for (int k0 = 0; k0 < IN_F; k0 += 4) {
        const int kk = k0 + 2 * hi;
        v2f a0 = *(const v2f*)(frow0 + kk);
        v2f a1 = *(const v2f*)(frow1 + kk);
#pragma unroll
        for (int nt = 0; nt < 4; ++nt) {
            v2f b = *(const v2f*)(ldsW1t + (nt * 16 + lo) * IN_F + kk);
            acc[0][nt] = __builtin_amdgcn_wmma_f32_16x16x4_f32(
                false, a0, false, b, (short)0, acc[0][nt], false, false);
            acc[1][nt] = __builtin_amdgcn_wmma_f32_16x16x4_f32(
                false, a1, false, b, (short)0, acc[1][nt], false, false);
        }
    }

    // ---- bias + ReLU, stage h tile (32x64) into this wave's LDS region.
    // D layout: VGPR v -> M = v + 8*hi (+16 for tile 1), N = lo.
    float* hW = ldsH + wave * (32 * HID_F);
#pragma unroll
    for (int mt = 0; mt < 2; ++mt)
#pragma unroll
        for (int v = 0; v < 8; ++v) {
            int m = mt * 16 + v + 8 * hi;
#pragma unroll
            for (int nt = 0; nt < 4; ++nt)
                hW[m * HID_F + nt * 16 + lo] =
                    fmaxf(acc[mt][nt][v] + b1[nt * 16 + lo], 0.f);
        }
    __syncthreads();

    // ---- GEMM2: [32 x 64] @ [64 x 32] -> 2 M-tiles x 2 N-tiles of 16x16
    v8f o[2][2];
#pragma unroll
    for (int mt = 0; mt < 2; ++mt)
#pragma unroll
        for (int nt = 0; nt < 2; ++nt) o[mt][nt] = v8f{0};

    for (int k0 = 0; k0 < HID_F; k0 += 4) {
        const int kk = k0 + 2 * hi;
        v2f a0 = *(const v2f*)(hW + lo * HID_F + kk);          // rows 0..15
        v2f a1 = *(const v2f*)(hW + (16 + lo) * HID_F + kk);   // rows 16..31
#pragma unroll
        for (int nt = 0; nt < 2; ++nt) {
            v2f b = *(const v2f*)(ldsW2t + (nt * 16 + lo) * HID_F + kk);
            o[0][nt] = __builtin_amdgcn_wmma_f32_16x16x4_f32(
                false, a0, false, b, (short)0, o[0][nt], false, false);
            o[1][nt] = __builtin_amdgcn_wmma_f32_16x16x4_f32(
                false, a1, false, b, (short)0, o[1][nt], false, false);
        }
    }

#pragma unroll
    for (int mt = 0; mt < 2; ++mt)
#pragma unroll
        for (int v = 0; v < 8; ++v) {
            int m = mt * 16 + v + 8 * hi;
            int rr = row0 + m;
            if (rr < N) {
                hh[(size_t)rr * OUT_F + lo]      = o[mt][0][v] + b2[lo];
                hh[(size_t)rr * OUT_F + 16 + lo] = o[mt][1][v] + b2[16 + lo];
            }
        }
}

// ---------------------------------------------------------------------------
// Graph-phase kernels (edge scatters, L2-resident node arrays)
// ---------------------------------------------------------------------------
__global__ void zero_f32(float* __restrict__ p, size_t n) {
    size_t i = (size_t)blockIdx.x * blockDim.x + threadIdx.x;
    if (i < n) p[i] = 0.f;
}

__global__ void deg_accum(const int* __restrict__ src, const int* __restrict__ dst,
                          float* __restrict__ deg, int E) {
    int e = blockIdx.x * blockDim.x + threadIdx.x;
    if (e < E) {
        atomicAdd(&deg[src[e]], 1.0f);
        atomicAdd(&deg[dst[e]], 1.0f);
    }
}

__global__ void make_dinv(const float* __restrict__ deg, float* __restrict__ dinv, int N) {
    int n = blockIdx.x * blockDim.x + threadIdx.x;
    if (n < N) dinv[n] = rsqrtf(deg[n] + 1.0f);   // self-loop degree
}

// ax[row] += (dinv[row]*dinv[col]) * x[col], both edge orientations.
// 8 lanes per directed edge, each lane handles 4 columns (float4 gather).
__global__ void ax_scatter(const int* __restrict__ src, const int* __restrict__ dst,
                           const float* __restrict__ dinv,
                           const float* __restrict__ hh,
                           float* __restrict__ ax, int E) {
    long idx = (long)blockIdx.x * blockDim.x + threadIdx.x;
    long de = idx >> 3;
    int  q  = (int)(idx & 7);
    if (de >= 2L * E) return;
    int r, c;
    if (de < E) { r = src[de]; c = dst[de]; }
    else        { r = dst[de - E]; c = src[de - E]; }
    float w = dinv[r] * dinv[c];
    float4 hv = ((const float4*)hh)[(size_t)c * 8 + q];
    float* p = ax + (size_t)r * OUT_F + q * 4;
    atomicAdd(p + 0, w * hv.x);
    atomicAdd(p + 1, w * hv.y);
    atomicAdd(p + 2, w * hv.z);
    atomicAdd(p + 3, w * hv.w);
}

// y = gamma*hh + (1-gamma)*(ax + dinv^2 * hh), in place over ax buffer.
__global__ void make_y(const float* __restrict__ hh, const float* __restrict__ dinv,
                       float* __restrict__ y, int N) {
    long i = (long)blockIdx.x * blockDim.x + threadIdx.x;
    if (i < (long)N * OUT_F) {
        int n = (int)(i >> 5);
        float dv = dinv[n];
        float h  = hh[i];
        y[i] = GAMMA_C * h + (1.f - GAMMA_C) * (y[i] + dv * dv * h);
    }
}

// One wave32 per edge, lane = output column (OUT_F == 32).
// z_e = l21proj(beta*(di*y[i] - dj*y[j])); acc += IncT contribution.
__global__ void z_scatter(const int* __restrict__ src, const int* __restrict__ dst,
                          const float* __restrict__ dinv,
                          const float* __restrict__ y,
                          float* __restrict__ acc, int E) {
    int gw = (blockIdx.x * blockDim.x + threadIdx.x) >> 5;
    int lane = threadIdx.x & 31;
    if (gw >= E) return;
    int s = src[gw], d = dst[gw];
    int i = s > d ? s : d;
    int j = s > d ? d : s;
    float di = dinv[i], dj = dinv[j];
    float zi = BETA_C * (di * y[(size_t)i * OUT_F + lane]
                       - dj * y[(size_t)j * OUT_F + lane]);
    float t = zi * zi;
#pragma unroll
    for (int m = 16; m >= 1; m >>= 1) t += __shfl_xor(t, m, 32);
    float rn = sqrtf(t);
    float scale = fminf(rn, LAMBDA1_C) / (rn > 0.f ? rn : 1.f);
    float zv = zi * scale;
    atomicAdd(&acc[(size_t)i * OUT_F + lane],  di * zv);
    atomicAdd(&acc[(size_t)j * OUT_F + lane], -dj * zv);
}

// x = y - gamma*acc, then log_softmax across 32 columns (one wave per node).
__global__ void final_logsoftmax(const float* __restrict__ y,
                                 const float* __restrict__ acc,
                                 float* __restrict__ out, int N) {
    int gw = (blockIdx.x * blockDim.x + threadIdx.x) >> 5;
    int lane = threadIdx.x & 31;
    if (gw >= N) return;
    size_t base = (size_t)gw * OUT_F;
    float v = y[base + lane] - GAMMA_C * acc[base + lane];
    float m = v;
#pragma unroll
    for (int k = 16; k >= 1; k >>= 1) m = fmaxf(m, __shfl_xor(m, k, 32));
    float ex = expf(v - m);
    float ss = ex;
#pragma unroll
    for (int k = 16; k >= 1; k >>= 1) ss += __shfl_xor(ss, k, 32);
    out[base + lane] = (v - m) - logf(ss);
}

// ---------------------------------------------------------------------------
extern "C" void kernel_launch(void* const* d_in, const int* in_sizes, int n_in,
                              void* d_out, int out_size, void* d_ws, size_t ws_size,
                              hipStream_t stream) {
    const float* feat = (const float*)d_in[0];
    const float* W1   = (const float*)d_in[1];
    const float* b1   = (const float*)d_in[2];
    const float* W2   = (const float*)d_in[3];
    const float* b2   = (const float*)d_in[4];
    const int*   esrc = (const int*)d_in[5];
    const int*   edst = (const int*)d_in[6];
    float* out = (float*)d_out;

    const int N = in_sizes[0] / IN_F;
    const int E = in_sizes[5];

    // Workspace layout (floats): hh[N*32] | y/ax[N*32] | acc[N*32] | deg[N] | dinv[N]
    float* hh   = (float*)d_ws;
    float* yb   = hh  + (size_t)N * OUT_F;
    float* acc  = yb  + (size_t)N * OUT_F;
    float* deg  = acc + (size_t)N * OUT_F;
    float* dinv = deg + N;

    // zero ax(=yb), acc, deg — contiguous region
    size_t zn = (size_t)N * (2 * OUT_F) + (size_t)N;
    zero_f32<<<(unsigned)((zn + 255) / 256), 256, 0, stream>>>(yb, zn);

    size_t shmem = (size_t)(IN_F * HID_F + HID_F * OUT_F + 8 * 32 * HID_F) * sizeof(float);
    mlp_wmma<<<(N + 255) / 256, 256, shmem, stream>>>(feat, W1, b1, W2, b2, hh, N);

    deg_accum<<<(E + 255) / 256, 256, 0, stream>>>(esrc, edst, deg, E);
    make_dinv<<<(N + 255) / 256, 256, 0, stream>>>(deg, dinv, N);

    long axItems = 16L * E;   // 2E directed edges * 8 lanes
    ax_scatter<<<(unsigned)((axItems + 255) / 256), 256, 0, stream>>>(esrc, edst, dinv, hh, yb, E);

    long yItems = (long)N * OUT_F;
    make_y<<<(unsigned)((yItems + 255) / 256), 256, 0, stream>>>(hh, dinv, yb, N);

    z_scatter<<<(E + 7) / 8, 256, 0, stream>>>(esrc, edst, dinv, yb, acc, E);

    final_logsoftmax<<<(N + 7) / 8, 256, 0, stream>>>(yb, acc, out, N);
}